// TorchLeavesLayer_26723286515898
// MI455X (gfx1250) — compile-verified
//
#include <hip/hip_runtime.h>
#include <cstdint>

// Problem constants (match reference)
#define B_ROWS   16384
#define F_TOT    256      // 128 cont + 128 cat features per row of x
#define NG       1024
#define NC       1024
#define NOUT     2048     // NG + NC output columns
#define KCAT     32
#define LOG_2PI  1.8378770664093453f

// Tiling
#define ROWS     64       // batch rows per block
#define COLS     256      // output columns per block == blockDim.x

// out[b, col] for col<NG:  -0.5*z^2 - log(std) - 0.5*log(2pi),  z=(x[b,gscope[col]]-mean)/std
// out[b, col] for col>=NG: cat_logp[c, int(x[b, cscope[c]])],    c = col-NG
//
// Bandwidth-bound (128 MB write-once store stream => ~5.5us floor @ 23.3 TB/s).
// CDNA5 paths: async global->LDS DMA (ASYNCcnt) for the x tile, LDS gathers
// (cat_logp slab stored transposed for conflict-free data-dependent gathers),
// non-temporal coalesced output stores, wave32-sized 256-thread blocks.

__global__ __launch_bounds__(COLS)
void spn_leaves_kernel(const float* __restrict__ x,
                       const float* __restrict__ gmean,
                       const float* __restrict__ gstd,
                       const float* __restrict__ clogp,
                       const int*   __restrict__ gscope,
                       const int*   __restrict__ cscope,
                       float*       __restrict__ out)
{
    __shared__ float s_x[ROWS * 128];    // 32 KB: one half (cont OR cat) of 64 x-rows
    __shared__ float s_lp[KCAT * COLS];  // 32 KB: cat_logp slab TRANSPOSED: [idx][col]

    const int  t      = threadIdx.x;
    const int  ct     = blockIdx.x;          // column tile 0..7 (0..3 gauss, 4..7 cat)
    const int  b0     = blockIdx.y * ROWS;   // first batch row of this tile
    const bool is_cat = (ct >= 4);
    const int  half   = is_cat ? 1 : 0;      // which 128-feature half of x we need

    // ---- Stage x[b0:b0+ROWS, half*128 : half*128+128] into LDS via async DMA ----
    {
        const uint64_t gbase = (uint64_t)(uintptr_t)x
                             + (uint64_t)b0 * (F_TOT * 4)
                             + (uint64_t)(half * 128 * 4);
        // 64 rows * 128 floats = 2048 16-byte chunks; 8 b128 DMAs per thread.
        #pragma unroll
        for (int i = 0; i < (ROWS * 128 / 4) / COLS; ++i) {
            const int      chunk = i * COLS + t;           // 0..2047
            const int      row   = chunk >> 5;             // 32 chunks per x-row
            const int      quad  = chunk & 31;
            const uint32_t voff  = (uint32_t)(row * (F_TOT * 4) + quad * 16);
            const uint32_t laddr = (uint32_t)(uintptr_t)(&s_x[chunk * 4]);
            asm volatile("global_load_async_to_lds_b128 %0, %1, %2 offset:0"
                         :: "v"(laddr), "v"(voff), "s"(gbase) : "memory");
        }
    }
    // ---- Cat tiles stage cat_logp[c0:c0+256, 0:32] TRANSPOSED into LDS ----
    // Transposed layout s_lp[k*COLS + j]: gather read addr = k*256 + t, and
    // 256 % 64 == 0, so the bank is t%64 -> conflict-free for any data idx.
    if (is_cat) {
        const int c0 = (ct - 4) * COLS;
        const float4* gp = (const float4*)(clogp + (size_t)c0 * KCAT);
        #pragma unroll
        for (int i = 0; i < (COLS * KCAT / 4) / COLS; ++i) {
            const int chunk = i * COLS + t;                // 16B chunk id, 0..2047
            const int j     = chunk >> 3;                  // column within tile
            const int k0    = (chunk & 7) * 4;             // first of 4 idx values
            const float4 v  = gp[chunk];                   // coalesced global read
            s_lp[(k0 + 0) * COLS + j] = v.x;               // one-time transposed
            s_lp[(k0 + 1) * COLS + j] = v.y;               // ds stores (cheap)
            s_lp[(k0 + 2) * COLS + j] = v.z;
            s_lp[(k0 + 3) * COLS + j] = v.w;
        }
    }
    asm volatile("s_wait_asynccnt 0" ::: "memory");  // wait for this wave's DMAs
    __syncthreads();                                  // make LDS visible to all waves

    const int col = ct * COLS + t;                    // 0..2047 output column
    float* op = out + (size_t)b0 * NOUT + col;        // row loop strides by NOUT

    if (!is_cat) {
        // -------- Gaussian leaf columns --------
        const int   g   = col;
        const int   f   = gscope[g];                  // feature in [0,128)
        const float m   = gmean[g];
        const float sd  = gstd[g];
        const float inv = 1.0f / sd;
        const float c0  = -logf(sd) - 0.5f * LOG_2PI;
        #pragma unroll 8
        for (int r = 0; r < ROWS; ++r) {
            const float xv = s_x[r * 128 + f];        // LDS gather
            const float z  = (xv - m) * inv;
            const float v  = fmaf(-0.5f * z, z, c0);
            __builtin_nontemporal_store(v, op + (size_t)r * NOUT);
        }
    } else {
        // -------- Categorical leaf columns --------
        const int c  = col - NG;
        const int fl = cscope[c] - 128;               // local index into cat half
        #pragma unroll 8
        for (int r = 0; r < ROWS; ++r) {
            const float xv  = s_x[r * 128 + fl];      // LDS gather
            const int   idx = (int)xv;                // values are exact ints 0..31
            const float v   = s_lp[idx * COLS + t];   // conflict-free LDS gather
            __builtin_nontemporal_store(v, op + (size_t)r * NOUT);
        }
    }
}

extern "C" void kernel_launch(void* const* d_in, const int* in_sizes, int n_in,
                              void* d_out, int out_size, void* d_ws, size_t ws_size,
                              hipStream_t stream) {
    const float* x      = (const float*)d_in[0];
    const float* gmean  = (const float*)d_in[1];
    const float* gstd   = (const float*)d_in[2];
    const float* clogp  = (const float*)d_in[3];
    const int*   gscope = (const int*)d_in[4];
    const int*   cscope = (const int*)d_in[5];

    dim3 grid(NOUT / COLS, B_ROWS / ROWS, 1);   // (8, 256)
    spn_leaves_kernel<<<grid, COLS, 0, stream>>>(x, gmean, gstd, clogp,
                                                 gscope, cscope, (float*)d_out);
}